// Nonlocal_Linear_48507360641757
// MI455X (gfx1250) — compile-verified
//
#include <hip/hip_runtime.h>
#include <hip/hip_bf16.h>
#include <math.h>

// Problem constants from the reference
#define N_  4
#define C_  64
#define H_  56
#define W_  56
#define HW  3136          // H_*W_
#define CR  16            // c_red = 0.25*C

typedef __attribute__((ext_vector_type(2))) float v2f;
typedef __attribute__((ext_vector_type(8))) float v8f;
typedef int v4i_ __attribute__((vector_size(16)));   // async-to-LDS payload type

// ---------------------------------------------------------------------------
// f_gen: f[n, ab, cd] = sum_{i<16} l[n,i,ab] * l[n,i,cd]
// A = theta^T (3136x16), B = theta (16x3136) read as B^T rows -> same gather.
// One wave computes a 16(ab) x 64(cd) strip; 4 waves/block -> 64x64 tile.
// ---------------------------------------------------------------------------
__global__ void __launch_bounds__(128)
f_gen_kernel(const float* __restrict__ l, float* __restrict__ f) {
    const int n     = blockIdx.z;
    const int wave  = threadIdx.x >> 5;
    const int lane  = threadIdx.x & 31;
    const int col   = lane & 15;
    const int khalf = (lane >> 4) << 1;          // 0 or 2 (K offset within 4-step)
    const int ab_base  = (blockIdx.y * 4 + wave) * 16;
    const int cd_base0 = blockIdx.x * 64;
    const float* lp = l + (size_t)n * C_ * HW;

    // A tiles for this wave's ab row: A[lane][v] = theta[n, k, ab_base+col]
    v2f a[4];
#pragma unroll
    for (int ks = 0; ks < 4; ++ks) {
        int k = ks * 4 + khalf;
        a[ks].x = lp[(size_t)k * HW + ab_base + col];
        a[ks].y = lp[(size_t)(k + 1) * HW + ab_base + col];
    }

#pragma unroll
    for (int t = 0; t < 4; ++t) {
        const int cd_base = cd_base0 + t * 16;
        v8f acc = {};
#pragma unroll
        for (int ks = 0; ks < 4; ++ks) {
            int k = ks * 4 + khalf;
            v2f b;
            b.x = lp[(size_t)k * HW + cd_base + col];
            b.y = lp[(size_t)(k + 1) * HW + cd_base + col];
            acc = __builtin_amdgcn_wmma_f32_16x16x4_f32(
                false, a[ks], false, b, (short)0, acc, false, false);
        }
        float* frow = f + (size_t)n * HW * HW
                        + (size_t)(ab_base + ((lane >> 4) << 3)) * HW
                        + cd_base + col;
#pragma unroll
        for (int v = 0; v < 8; ++v)
            frow[(size_t)v * HW] = acc[v];
    }
}

// ---------------------------------------------------------------------------
// gemm_abt64: D[n, m, j] = scale * sum_k A[n, m, k] * B[n, j, k]
// M = 64 (4 tiles), K = HW, j-dim tiled 64 per block (one 16-wide tile/wave).
// Used for: g = l @ W_lin^T     (bStride = 0, W_lin shared across n)
//           out = g @ f^T       (bStride = HW*HW)
// ---------------------------------------------------------------------------
__global__ void __launch_bounds__(128)
gemm_abt64_kernel(const float* __restrict__ A, size_t aStride,
                  const float* __restrict__ B, size_t bStride,
                  float* __restrict__ D, size_t dStride, float scale) {
    const int n     = blockIdx.y;
    const int wave  = threadIdx.x >> 5;
    const int lane  = threadIdx.x & 31;
    const int col   = lane & 15;
    const int khalf = (lane >> 4) << 1;
    const int j     = blockIdx.x * 64 + wave * 16 + col;   // output column

    const float* Ab   = A + (size_t)n * aStride;
    const float* Brow = B + (size_t)n * bStride + (size_t)j * HW;

    v8f acc[4] = {};
    for (int k0 = 0; k0 < HW; k0 += 16) {
        __builtin_prefetch(Brow + k0 + 128);               // global_prefetch_b8
#pragma unroll
        for (int ku = 0; ku < 16; ku += 4) {
            const int kk = k0 + ku + khalf;
            const v2f b = *(const v2f*)(Brow + kk);
#pragma unroll
            for (int mt = 0; mt < 4; ++mt) {
                const v2f a = *(const v2f*)(Ab + (size_t)(mt * 16 + col) * HW + kk);
                acc[mt] = __builtin_amdgcn_wmma_f32_16x16x4_f32(
                    false, a, false, b, (short)0, acc[mt], false, false);
            }
        }
    }

    float* Dn = D + (size_t)n * dStride;
#pragma unroll
    for (int mt = 0; mt < 4; ++mt)
#pragma unroll
        for (int v = 0; v < 8; ++v) {
            const int m = mt * 16 + v + ((lane >> 4) << 3);
            Dn[(size_t)m * HW + j] = acc[mt][v] * scale;
        }
}

// ---------------------------------------------------------------------------
// softmax over axis=1 (ab) for each (n, cd): column softmax, online max/sum.
// Consecutive threads take consecutive columns -> coalesced row reads.
// ---------------------------------------------------------------------------
__global__ void __launch_bounds__(256)
softmax_ab_kernel(float* __restrict__ f) {
    const int n   = blockIdx.y;
    const int col = blockIdx.x * 256 + threadIdx.x;
    if (col >= HW) return;
    float* base = f + (size_t)n * HW * HW + col;

    float m = -3.4e38f, s = 0.f;
    for (int ab = 0; ab < HW; ++ab) {
        const float x  = base[(size_t)ab * HW];
        const float mn = fmaxf(m, x);
        s = s * __expf(m - mn) + __expf(x - mn);
        m = mn;
    }
    const float inv = 1.f / s;
    for (int ab = 0; ab < HW; ++ab) {
        const size_t o = (size_t)ab * HW;
        base[o] = __expf(base[o] - m) * inv;
    }
}

// ---------------------------------------------------------------------------
// softmax over axis=2 (cd) for each (n, ab): one row (12.5KB) staged in LDS.
// On gfx1250 the staging uses GLOBAL_LOAD_ASYNC_TO_LDS_B128 (ASYNCcnt path):
// each wave waits on its own async copies (s_wait_asynccnt 0), then a
// workgroup barrier makes all copies visible before the reduction.
// wave32 shuffle + LDS block reduction.
// ---------------------------------------------------------------------------
__global__ void __launch_bounds__(256)
softmax_cd_kernel(float* __restrict__ f) {
    __shared__ __align__(16) float buf[HW];
    __shared__ float r1[8], r2[8];
    const int n = blockIdx.y, row = blockIdx.x, tid = threadIdx.x;
    const int wave = tid >> 5, lane = tid & 31;
    float* base = f + (size_t)n * HW * HW + (size_t)row * HW;

#if defined(__gfx1250__) && \
    __has_builtin(__builtin_amdgcn_global_load_async_to_lds_b128) && \
    __has_builtin(__builtin_amdgcn_s_wait_asynccnt)
    // HW == 784 * 4: the row is exactly 784 16-byte chunks (rows of f are
    // 16B-aligned: HW*4 = 12544 bytes is a multiple of 16).
    for (int j4 = tid; j4 < HW / 4; j4 += 256) {
        __builtin_amdgcn_global_load_async_to_lds_b128(
            (__attribute__((address_space(1))) v4i_*)(base + j4 * 4),
            (__attribute__((address_space(3))) v4i_*)(buf + j4 * 4),
            0, 0);
    }
    __builtin_amdgcn_s_wait_asynccnt(0);
    __syncthreads();
    float m = -3.4e38f;
    for (int jx = tid; jx < HW; jx += 256) m = fmaxf(m, buf[jx]);
#else
    float m = -3.4e38f;
    for (int jx = tid; jx < HW; jx += 256) {
        const float x = base[jx];
        buf[jx] = x;
        m = fmaxf(m, x);
    }
#endif
#pragma unroll
    for (int off = 16; off > 0; off >>= 1) m = fmaxf(m, __shfl_xor(m, off, 32));
    if (lane == 0) r1[wave] = m;
    __syncthreads();
    m = r1[0];
#pragma unroll
    for (int i = 1; i < 8; ++i) m = fmaxf(m, r1[i]);

    float s = 0.f;
    for (int jx = tid; jx < HW; jx += 256) s += __expf(buf[jx] - m);
#pragma unroll
    for (int off = 16; off > 0; off >>= 1) s += __shfl_xor(s, off, 32);
    if (lane == 0) r2[wave] = s;
    __syncthreads();
    s = 0.f;
#pragma unroll
    for (int i = 0; i < 8; ++i) s += r2[i];

    const float inv = 1.f / s;
    for (int jx = tid; jx < HW; jx += 256) base[jx] = __expf(buf[jx] - m) * inv;
}

// ---------------------------------------------------------------------------
// Depthwise 3x3 conv (pad 1) + per-block partial BN sums (deterministic).
// ---------------------------------------------------------------------------
__global__ void __launch_bounds__(256)
dwconv_kernel(const float* __restrict__ in, const float* __restrict__ kern,
              float* __restrict__ out, float* __restrict__ partial) {
    const int bx = blockIdx.x, c = blockIdx.y, n = blockIdx.z;
    const int tid = threadIdx.x;
    const int p = bx * 256 + tid;
    const float* plane = in + ((size_t)n * C_ + c) * HW;

    float v = 0.f;
    if (p < HW) {
        const int h = p / W_, w = p % W_;
#pragma unroll
        for (int ky = 0; ky < 3; ++ky) {
            const int hh = h + ky - 1;
            if (hh < 0 || hh >= H_) continue;
#pragma unroll
            for (int kx = 0; kx < 3; ++kx) {
                const int ww = w + kx - 1;
                if (ww < 0 || ww >= W_) continue;
                v += plane[hh * W_ + ww] * kern[c * 9 + ky * 3 + kx];
            }
        }
        out[((size_t)n * C_ + c) * HW + p] = v;
    }

    float v2 = v * v;
#pragma unroll
    for (int off = 16; off > 0; off >>= 1) {
        v  += __shfl_xor(v,  off, 32);
        v2 += __shfl_xor(v2, off, 32);
    }
    __shared__ float s1[8], s2[8];
    const int wave = tid >> 5, lane = tid & 31;
    if (lane == 0) { s1[wave] = v; s2[wave] = v2; }
    __syncthreads();
    if (tid == 0) {
        float a = 0.f, b = 0.f;
#pragma unroll
        for (int i = 0; i < 8; ++i) { a += s1[i]; b += s2[i]; }
        const size_t pidx = ((size_t)c * (N_ * 13) + (size_t)n * 13 + bx) * 2;
        partial[pidx]     = a;
        partial[pidx + 1] = b;
    }
}

__global__ void stats_reduce_kernel(const float* __restrict__ partial,
                                    float* __restrict__ stats) {
    const int c = threadIdx.x;
    if (c >= C_) return;
    float a = 0.f, b = 0.f;
    for (int j = 0; j < N_ * 13; ++j) {
        a += partial[((size_t)c * (N_ * 13) + j) * 2];
        b += partial[((size_t)c * (N_ * 13) + j) * 2 + 1];
    }
    stats[c]      = a;
    stats[C_ + c] = b;
}

// ---------------------------------------------------------------------------
// BatchNorm apply + residual:  out = (conv - mean)*rsqrt(var+eps)*w + b + l
// ---------------------------------------------------------------------------
__global__ void __launch_bounds__(256)
bn_apply_kernel(const float* __restrict__ conv, const float* __restrict__ stats,
                const float* __restrict__ bnw, const float* __restrict__ bnb,
                const float* __restrict__ l, float* __restrict__ out) {
    const int idx = blockIdx.x * 256 + threadIdx.x;
    if (idx >= N_ * C_ * HW) return;
    const int c = (idx / HW) % C_;
    const float cnt  = (float)(N_ * HW);
    const float mean = stats[c] / cnt;
    const float var  = stats[C_ + c] / cnt - mean * mean;
    const float sc   = rsqrtf(var + 1e-5f) * bnw[c];
    out[idx] = (conv[idx] - mean) * sc + bnb[c] + l[idx];
}

// ---------------------------------------------------------------------------
extern "C" void kernel_launch(void* const* d_in, const int* in_sizes, int n_in,
                              void* d_out, int out_size, void* d_ws, size_t ws_size,
                              hipStream_t stream) {
    const float* l      = (const float*)d_in[0];   // (N,C,H,W)
    const float* W_lin  = (const float*)d_in[1];   // (hw,hw)
    const float* dw_k   = (const float*)d_in[2];   // (C,1,3,3)
    const float* bnw    = (const float*)d_in[3];   // (C,)
    const float* bnb    = (const float*)d_in[4];   // (C,)
    float* out          = (float*)d_out;

    // Workspace layout (floats): f | g | t1(attn out) | t2(conv out) | partials | stats
    float* ws = (float*)d_ws;
    const size_t nchw = (size_t)N_ * C_ * HW;
    float* f       = ws;                                   // N*HW*HW  (~157 MB)
    float* g       = f  + (size_t)N_ * HW * HW;            // N*C*HW
    float* t1      = g  + nchw;                            // attention output
    float* t2      = t1 + nchw;                            // conv output
    float* partial = t2 + nchw;                            // C * N*13 * 2
    float* stats   = partial + (size_t)C_ * N_ * 13 * 2;   // 2*C

    // 1. f = theta^T @ theta      (WMMA f32 16x16x4, K=16)
    f_gen_kernel<<<dim3(HW / 64, HW / 64, N_), 128, 0, stream>>>(l, f);

    // 2. g = l @ W_lin^T          (WMMA, K=3136; W_lin shared across batch)
    gemm_abt64_kernel<<<dim3(HW / 64, N_), 128, 0, stream>>>(
        l, (size_t)C_ * HW, W_lin, (size_t)0, g, (size_t)C_ * HW, 1.0f);

    // 3. softmax over ab (axis=1), then over cd (axis=2)
    softmax_ab_kernel<<<dim3((HW + 255) / 256, N_), 256, 0, stream>>>(f);
    softmax_cd_kernel<<<dim3(HW, N_), 256, 0, stream>>>(f);

    // 4. t1 = g @ f^T, scaled by W/H (== 1.0)
    gemm_abt64_kernel<<<dim3(HW / 64, N_), 128, 0, stream>>>(
        g, (size_t)C_ * HW, f, (size_t)HW * HW, t1, (size_t)C_ * HW,
        (float)W_ / (float)H_);

    // 5. depthwise conv + partial BN stats; deterministic channel reduce
    dwconv_kernel<<<dim3(13, C_, N_), 256, 0, stream>>>(t1, dw_k, t2, partial);
    stats_reduce_kernel<<<1, 64, 0, stream>>>(partial, stats);

    // 6. BN apply + residual
    bn_apply_kernel<<<dim3((unsigned)((nchw + 255) / 256)), 256, 0, stream>>>(
        t2, stats, bnw, bnb, l, out);
}